// TrueLayerEnsembleMLP_910533067014
// MI455X (gfx1250) — compile-verified
//
#include <hip/hip_runtime.h>

typedef __attribute__((ext_vector_type(2))) float v2f;
typedef __attribute__((ext_vector_type(8))) float v8f;

#define BM 128
#define BN 128
#define BK 32
#define LDA 36    // BK + 4 pad: 16B-aligned b128 async stores, conflict-free A-frag reads
#define LDBT 34   // BK + 2 pad: K-innermost B tile; 8B-aligned b64 frag reads, conflict-free
#define KENS 8
#define ASZ (BM * LDA)    // 4608 floats per A buffer
#define BSZ (BN * LDBT)   // 4352 floats per B buffer
// double-buffered LDS: 2*(ASZ+BSZ)*4 = 71680 B of the 320 KB WGP pool

// Generic (flat) address of an LDS object: low 32 bits are the LDS byte offset
__device__ __forceinline__ unsigned lds_off(const void* p) {
    return (unsigned)(size_t)p;
}

// CDNA5 async memory->LDS copies, GVS (saddr) mode: uniform SGPR64 base +
// 32-bit per-lane byte offset. ASYNCcnt-tracked, no VGPR round-trip.
__device__ __forceinline__ void async_b128(unsigned lds, unsigned voff, const float* base) {
    asm volatile("global_load_async_to_lds_b128 %0, %1, %2"
                 :: "v"(lds), "v"(voff), "s"((unsigned long long)base) : "memory");
}
__device__ __forceinline__ void async_b32(unsigned lds, unsigned voff, const float* base) {
    asm volatile("global_load_async_to_lds_b32 %0, %1, %2"
                 :: "v"(lds), "v"(voff), "s"((unsigned long long)base) : "memory");
}
__device__ __forceinline__ void wait_async0() {
    asm volatile("s_wait_asynccnt 0x0" ::: "memory");
}

template <int K, int N, int RELU>
__global__ __launch_bounds__(256) void ens_gemm(
    const float* __restrict__ A,     // [M, K] activations, row-major
    const float* __restrict__ W,     // [K, N, KENS] ensemble weights
    const float* __restrict__ Bias,  // [N, KENS]
    const int*   __restrict__ index, // [3]
    int layer,
    float* __restrict__ Out)         // [M, N]
{
    __shared__ __align__(16) float smem[2 * ASZ + 2 * BSZ];

    const int tid  = threadIdx.x;
    const int lane = tid & 31;
    const int wave = tid >> 5;
    const int lq   = lane & 15;     // M/N index within a 16x16 tile
    const int half = lane >> 4;     // selects K pair (A/B) or row half (C/D)

    const int blockN0 = blockIdx.x * BN;
    const int blockM0 = blockIdx.y * BM;

    const int waveM = (wave & 3) * 32;   // 4 waves tile M (32 rows each)
    const int waveN = (wave >> 2) * 64;  // 2 waves tile N (64 cols each)

    const int idx = index[layer];        // uniform scalar load
    const float* Wsel = W + idx;         // uniform: ensemble-selected base (SGPR)

    v8f acc[2][4];
    const v8f vzero = {0.f, 0.f, 0.f, 0.f, 0.f, 0.f, 0.f, 0.f};
#pragma unroll
    for (int i = 0; i < 2; ++i)
#pragma unroll
        for (int j = 0; j < 4; ++j)
            acc[i][j] = vzero;

    // A tile: 128 rows x 8 float4 cols; 4 async b128 per thread
    const int a_c4 = (tid & 7) * 4;
    const int a_r  = tid >> 3;           // 0..31
    // B tile gather: 128 cols x 32 rows; 16 async b32 per thread
    const int b_n = tid & 127;
    const int b_k = tid >> 7;            // 0..1

    // Per-lane 32-bit byte offsets (bases advance by k0 each tile)
    const unsigned aOff0 = (unsigned)(((blockM0 + a_r) * K + a_c4) * 4);
    const unsigned bOff0 = (unsigned)(((b_k * N) + (blockN0 + b_n)) * KENS * 4);

    auto issue_tile = [&](int k0, int buf) {
        const unsigned aBase = buf * ASZ;
        const unsigned bBase = 2 * ASZ + buf * BSZ;
        const unsigned aK = aOff0 + (unsigned)(k0 * 4);
        const unsigned bK = bOff0 + (unsigned)(k0 * N * KENS * 4);
#pragma unroll
        for (int i = 0; i < 4; ++i) {
            int row = a_r + 32 * i;
            async_b128(lds_off(&smem[aBase + row * LDA + a_c4]),
                       aK + (unsigned)(32 * i * K * 4), A);
        }
#pragma unroll
        for (int kk = 0; kk < BK; kk += 2) {
            int k = b_k + kk;
            // ensemble-strided gather W[(k, n, idx)] -> K-innermost LDS tile
            async_b32(lds_off(&smem[bBase + b_n * LDBT + k]),
                      bK + (unsigned)(kk * N * KENS * 4), Wsel);
        }
    };

    issue_tile(0, 0);
    wait_async0();
    __syncthreads();

    const int nIter = K / BK;
    for (int it = 0; it < nIter; ++it) {
        const int cur = it & 1;
        if (it + 1 < nIter)
            issue_tile((it + 1) * BK, cur ^ 1);   // prefetch next tile into other buffer

        const unsigned aBase = cur * ASZ;
        const unsigned bBase = 2 * ASZ + cur * BSZ;
#pragma unroll
        for (int k4 = 0; k4 < BK; k4 += 4) {
            const int kf = k4 + half * 2;   // this lane's K pair base
            v2f af[2];
#pragma unroll
            for (int ms = 0; ms < 2; ++ms)
                af[ms] = *reinterpret_cast<const v2f*>(
                    &smem[aBase + (waveM + ms * 16 + lq) * LDA + kf]);
            v2f bf[4];
#pragma unroll
            for (int ns = 0; ns < 4; ++ns)
                bf[ns] = *reinterpret_cast<const v2f*>(
                    &smem[bBase + (waveN + ns * 16 + lq) * LDBT + kf]);
#pragma unroll
            for (int ms = 0; ms < 2; ++ms)
#pragma unroll
                for (int ns = 0; ns < 4; ++ns)
                    acc[ms][ns] = __builtin_amdgcn_wmma_f32_16x16x4_f32(
                        /*neg_a=*/false, af[ms], /*neg_b=*/false, bf[ns],
                        /*c_mod=*/(short)0, acc[ms][ns],
                        /*reuse_a=*/false, /*reuse_b=*/false);
        }
        wait_async0();      // our async copies into the next buffer are done in LDS
        __syncthreads();    // everyone's are; current buffer free for reuse
    }

    // Epilogue: ensemble-selected bias + optional ReLU
#pragma unroll
    for (int ns = 0; ns < 4; ++ns) {
        int col = blockN0 + waveN + ns * 16 + lq;
        float bias = Bias[(size_t)col * KENS + idx];
#pragma unroll
        for (int ms = 0; ms < 2; ++ms) {
            int rowBase = blockM0 + waveM + ms * 16 + half * 8;
#pragma unroll
            for (int v = 0; v < 8; ++v) {
                float val = acc[ms][ns][v] + bias;
                if (RELU) val = fmaxf(val, 0.0f);
                Out[(size_t)(rowBase + v) * N + col] = val;
            }
        }
    }
}

extern "C" void kernel_launch(void* const* d_in, const int* in_sizes, int n_in,
                              void* d_out, int out_size, void* d_ws, size_t ws_size,
                              hipStream_t stream) {
    (void)in_sizes; (void)n_in; (void)out_size; (void)ws_size;
    const float* x     = (const float*)d_in[0];
    const int*   index = (const int*)  d_in[1];
    const float* w0    = (const float*)d_in[2];
    const float* b0    = (const float*)d_in[3];
    const float* w1    = (const float*)d_in[4];
    const float* b1    = (const float*)d_in[5];
    const float* w2    = (const float*)d_in[6];
    const float* b2    = (const float*)d_in[7];
    float* out = (float*)d_out;

    float* h1 = (float*)d_ws;                  // [1024, 2048] fp32
    float* h2 = h1 + (size_t)1024 * 2048;      // [1024, 2048] fp32

    dim3 blk(256);
    ens_gemm<2048, 2048, 1><<<dim3(2048 / BN, 1024 / BM), blk, 0, stream>>>(
        x,  w0, b0, index, 0, h1);
    ens_gemm<2048, 2048, 1><<<dim3(2048 / BN, 1024 / BM), blk, 0, stream>>>(
        h1, w1, b1, index, 1, h2);
    ens_gemm<2048, 1024, 0><<<dim3(1024 / BN, 1024 / BM), blk, 0, stream>>>(
        h2, w2, b2, index, 2, out);
}